// Bayesian_MuRP_32384053411871
// MI455X (gfx1250) — compile-verified
//
#include <hip/hip_runtime.h>
#include <hip/hip_bf16.h>

typedef float v2f __attribute__((ext_vector_type(2)));
typedef float v8f __attribute__((ext_vector_type(8)));

#define B_    2048
#define S_    128
#define H_    1024
#define LAT_  50
#define HID_  300
#define MD_   100
#define NR_   4
#define EPAD_ 128            // MD padded to power of two
#define ESTR_ 256            // 2*EPAD_
#define ZSTR_ 64             // LAT padded
#define TPW_  800            // 2*NR*MD
#define BN_EPS_ 1e-5f

#define LOGITS_OFF 0
#define MU_OFF  (B_*NR_)                 // 8192
#define LV_OFF  (MU_OFF + B_*LAT_)       // 110592
#define REL_OFF (LV_OFF + B_*LAT_)       // 212992

#define WMMA_F32(a, b, c) \
  __builtin_amdgcn_wmma_f32_16x16x4_f32(false, (a), false, (b), (short)0, (c), false, false)

__device__ __forceinline__ float wave_sum(float v) {
#pragma unroll
  for (int off = 16; off; off >>= 1) v += __shfl_xor(v, off, 32);
  return v;
}

// ---------------------------------------------------------------------------
// K1: e = gather(lm, epos) @ w_event + b_event  ->  E[B,256] (zero-padded)
// One wave per 16x32 tile (2 accumulators, A reused). 128 Mtiles x 4 Npairs x 2.
// Loads are unconditional (clamped addresses); pad columns zeroed at store.
// ---------------------------------------------------------------------------
__global__ __launch_bounds__(256) void k_event(const float* __restrict__ lm,
    const int* __restrict__ ep1, const int* __restrict__ ep2,
    const float* __restrict__ w, const float* __restrict__ bias,
    float* __restrict__ E) {
  const int wid  = (blockIdx.x * blockDim.x + threadIdx.x) >> 5;
  const int lane = threadIdx.x & 31;
  const int hf   = lane >> 4;
  const int l    = lane & 15;
  if (wid >= 128 * 8) return;
  const int mt    = wid >> 3;
  const int rem   = wid & 7;
  const int which = rem >> 2;
  const int n0    = (rem & 3) * 32;
  const int row   = mt * 16 + l;
  const int pos   = which ? ep2[row] : ep1[row];
  const float* arow = lm + ((size_t)row * S_ + (size_t)pos) * H_ + hf * 2;
  const int  n_a = n0 + l;
  const int  n_b = n_a + 16;
  const bool va  = (n_a < MD_);
  const bool vb  = (n_b < MD_);
  const int  ca  = va ? n_a : (MD_ - 1);     // clamped, always in-bounds
  const int  cb  = vb ? n_b : (MD_ - 1);
  const float* wp = w + (size_t)(hf * 2) * MD_;
  v8f acc0 = {0.f,0.f,0.f,0.f,0.f,0.f,0.f,0.f};
  v8f acc1 = {0.f,0.f,0.f,0.f,0.f,0.f,0.f,0.f};
  for (int k0 = 0; k0 < H_; k0 += 64) {
    __builtin_prefetch((const void*)(arow + k0 + 64), 0, 1);   // global_prefetch_b8
    const float* ap  = arow + k0;
    const float* bpa = wp + (size_t)k0 * MD_ + ca;
    const float* bpb = wp + (size_t)k0 * MD_ + cb;
#pragma unroll
    for (int j = 0; j < 64; j += 4) {
      v2f a;  a[0]  = ap[j];                  a[1]  = ap[j + 1];
      v2f b0; b0[0] = bpa[(size_t)j * MD_];   b0[1] = bpa[(size_t)(j + 1) * MD_];
      v2f b1; b1[0] = bpb[(size_t)j * MD_];   b1[1] = bpb[(size_t)(j + 1) * MD_];
      acc0 = WMMA_F32(a, b0, acc0);
      acc1 = WMMA_F32(a, b1, acc1);
    }
  }
  const float bs0 = bias[ca];
  const float bs1 = bias[cb];
  float* eb = E + (size_t)(which * EPAD_ + n0 + l);
#pragma unroll
  for (int i = 0; i < 8; ++i) {
    const int rr = mt * 16 + hf * 8 + i;
    eb[(size_t)rr * ESTR_ +  0] = va ? acc0[i] + bs0 : 0.f;  // pads exact 0
    eb[(size_t)rr * ESTR_ + 16] = vb ? acc1[i] + bs1 : 0.f;
  }
}

// ---------------------------------------------------------------------------
// K2: reps = dropout(relu(concat(e1,e2) @ w_fc1 + b_fc1))  -> reps[B,300]
// A = E[B,256] (pad cols are exact 0 => clamped B rows contribute nothing).
// kk<128 -> w row kk ; kk>=128 -> w row kk-28 (clamped <200).
// ---------------------------------------------------------------------------
__global__ __launch_bounds__(256) void k_fc1(const float* __restrict__ E,
    const float* __restrict__ w, const float* __restrict__ bias,
    const float* __restrict__ dropu, float* __restrict__ reps) {
  const int wid  = (blockIdx.x * blockDim.x + threadIdx.x) >> 5;
  const int lane = threadIdx.x & 31;
  const int hf   = lane >> 4;
  const int l    = lane & 15;
  if (wid >= 128 * 10) return;
  const int mt = wid / 10;
  const int pr = wid % 10;
  const int n0 = pr * 32;
  const int  n_a = n0 + l;
  const int  n_b = n_a + 16;
  const bool va  = (n_a < HID_);
  const bool vb  = (n_b < HID_);
  const int  ca  = va ? n_a : 0;
  const int  cb  = vb ? n_b : 0;
  const float* arow = E + (size_t)(mt * 16 + l) * ESTR_ + hf * 2;
  v8f acc0 = {0.f,0.f,0.f,0.f,0.f,0.f,0.f,0.f};
  v8f acc1 = {0.f,0.f,0.f,0.f,0.f,0.f,0.f,0.f};
  for (int k0 = 0; k0 < ESTR_; k0 += 64) {
    const float* ap = arow + k0;
#pragma unroll
    for (int j = 0; j < 64; j += 4) {
      v2f a; a[0] = ap[j]; a[1] = ap[j + 1];
      v2f b0, b1;
#pragma unroll
      for (int t = 0; t < 2; ++t) {
        const int kk = k0 + j + hf * 2 + t;
        int wr = (kk < EPAD_) ? kk : kk - (EPAD_ - MD_);
        wr = (wr < 2 * MD_) ? wr : 0;           // clamp; A is 0 there anyway
        const float* wrow = w + (size_t)wr * HID_;
        b0[t] = wrow[ca];
        b1[t] = wrow[cb];
      }
      acc0 = WMMA_F32(a, b0, acc0);
      acc1 = WMMA_F32(a, b1, acc1);
    }
  }
  const float bb0 = bias[ca];
  const float bb1 = bias[cb];
#pragma unroll
  for (int i = 0; i < 8; ++i) {
    const int rr = mt * 16 + hf * 8 + i;
    if (va) {
      float v = fmaxf(acc0[i] + bb0, 0.f);
      v *= (dropu[(size_t)rr * HID_ + n_a] > 0.5f) ? 2.f : 0.f;
      reps[(size_t)rr * HID_ + n_a] = v;
    }
    if (vb) {
      float v = fmaxf(acc1[i] + bb1, 0.f);
      v *= (dropu[(size_t)rr * HID_ + n_b] > 0.5f) ? 2.f : 0.f;
      reps[(size_t)rr * HID_ + n_b] = v;
    }
  }
}

// ---------------------------------------------------------------------------
// K3: mu_lin = reps@w_mu+b_mu ; sig_lin = reps@w_sigma+b_sigma  (stride 64)
// Pad outputs (cols 50..63) are written with clamped-duplicate values but
// never read downstream.
// ---------------------------------------------------------------------------
__global__ __launch_bounds__(256) void k_musig(const float* __restrict__ reps,
    const float* __restrict__ wmu, const float* __restrict__ bmu,
    const float* __restrict__ wsg, const float* __restrict__ bsg,
    float* __restrict__ mul, float* __restrict__ sgl) {
  const int wid  = (blockIdx.x * blockDim.x + threadIdx.x) >> 5;
  const int lane = threadIdx.x & 31;
  const int hf   = lane >> 4;
  const int l    = lane & 15;
  if (wid >= 128 * 4) return;
  const int mt  = wid >> 2;
  const int rem = wid & 3;
  const int sel = rem >> 1;
  const int n0  = (rem & 1) * 32;
  const int n_a = n0 + l;
  const int n_b = n_a + 16;
  const int ca  = (n_a < LAT_) ? n_a : (LAT_ - 1);
  const int cb  = (n_b < LAT_) ? n_b : (LAT_ - 1);
  const float* w  = sel ? wsg : wmu;
  const float* bi = sel ? bsg : bmu;
  float*       o  = sel ? sgl : mul;
  const float* ap = reps + (size_t)(mt * 16 + l) * HID_ + hf * 2;
  const float* wp = w + (size_t)(hf * 2) * LAT_;
  v8f acc0 = {0.f,0.f,0.f,0.f,0.f,0.f,0.f,0.f};
  v8f acc1 = {0.f,0.f,0.f,0.f,0.f,0.f,0.f,0.f};
#pragma unroll 5
  for (int k = 0; k < HID_; k += 4) {     // 300 % 4 == 0
    v2f a;  a[0]  = ap[k];                          a[1]  = ap[k + 1];
    v2f b0; b0[0] = wp[(size_t)k * LAT_ + ca];      b0[1] = wp[(size_t)(k + 1) * LAT_ + ca];
    v2f b1; b1[0] = wp[(size_t)k * LAT_ + cb];      b1[1] = wp[(size_t)(k + 1) * LAT_ + cb];
    acc0 = WMMA_F32(a, b0, acc0);
    acc1 = WMMA_F32(a, b1, acc1);
  }
  const float bb0 = bi[ca];
  const float bb1 = bi[cb];
#pragma unroll
  for (int i = 0; i < 8; ++i) {
    const int rr = mt * 16 + hf * 8 + i;
    o[(size_t)rr * ZSTR_ + n_a] = acc0[i] + bb0;
    o[(size_t)rr * ZSTR_ + n_b] = acc1[i] + bb1;
  }
}

// ---------------------------------------------------------------------------
// K4: batchnorm statistics over B for 100 columns (50 mu + 50 sigma)
// ---------------------------------------------------------------------------
__global__ __launch_bounds__(256) void k_bnstats(const float* __restrict__ mul,
    const float* __restrict__ sgl, float* __restrict__ bnm, float* __restrict__ bni) {
  const int c = blockIdx.x;               // 0..99
  const float* src = (c < LAT_) ? (mul + c) : (sgl + (c - LAT_));
  float s = 0.f, s2 = 0.f;
  for (int r = threadIdx.x; r < B_; r += 256) {
    const float x = src[(size_t)r * ZSTR_];
    s += x; s2 += x * x;
  }
  __shared__ float sh1[256], sh2[256];
  sh1[threadIdx.x] = s; sh2[threadIdx.x] = s2;
  __syncthreads();
  for (int off = 128; off; off >>= 1) {
    if (threadIdx.x < off) {
      sh1[threadIdx.x] += sh1[threadIdx.x + off];
      sh2[threadIdx.x] += sh2[threadIdx.x + off];
    }
    __syncthreads();
  }
  if (threadIdx.x == 0) {
    const float m   = sh1[0] / (float)B_;
    const float var = sh2[0] / (float)B_ - m * m;
    bnm[c] = m;
    bni[c] = rsqrtf(var + BN_EPS_);
  }
}

// ---------------------------------------------------------------------------
// K5: BN affine, write mu/logvar outputs, z = eps*sqrt(exp(lv)) + mu (pad 0)
// ---------------------------------------------------------------------------
__global__ __launch_bounds__(256) void k_zfin(const float* __restrict__ mul,
    const float* __restrict__ sgl, const float* __restrict__ bnm,
    const float* __restrict__ bni, const float* __restrict__ gmu,
    const float* __restrict__ bemu, const float* __restrict__ gva,
    const float* __restrict__ beva, const float* __restrict__ eps,
    float* __restrict__ out, float* __restrict__ z) {
  const int t = blockIdx.x * blockDim.x + threadIdx.x;
  const int b = t >> 6, j = t & 63;
  if (b >= B_) return;
  if (j < LAT_) {
    const float mu = (mul[(size_t)b * ZSTR_ + j] - bnm[j]) * bni[j] * gmu[j] + bemu[j];
    const float lv = (sgl[(size_t)b * ZSTR_ + j] - bnm[LAT_ + j]) * bni[LAT_ + j] * gva[j] + beva[j];
    out[MU_OFF + (size_t)b * LAT_ + j] = mu;
    out[LV_OFF + (size_t)b * LAT_ + j] = lv;
    z[(size_t)b * ZSTR_ + j] = eps[(size_t)b * LAT_ + j] * sqrtf(expf(lv)) + mu;
  } else {
    z[(size_t)b * ZSTR_ + j] = 0.f;      // pad => clamped B rows below are inert
  }
}

// ---------------------------------------------------------------------------
// K6: tp = tanh(z @ w_mure + b_mure)  -> tp[B,800]; z pad cols are exact 0.
// ---------------------------------------------------------------------------
__global__ __launch_bounds__(256) void k_mure(const float* __restrict__ z,
    const float* __restrict__ w, const float* __restrict__ bias,
    float* __restrict__ tp) {
  const int wid  = (blockIdx.x * blockDim.x + threadIdx.x) >> 5;
  const int lane = threadIdx.x & 31;
  const int hf   = lane >> 4;
  const int l    = lane & 15;
  if (wid >= 128 * 25) return;
  const int mt = wid / 25;
  const int pr = wid % 25;
  const int n_a = pr * 32 + l;           // < 800 always
  const int n_b = n_a + 16;
  const float* ap = z + (size_t)(mt * 16 + l) * ZSTR_ + hf * 2;
  v8f acc0 = {0.f,0.f,0.f,0.f,0.f,0.f,0.f,0.f};
  v8f acc1 = {0.f,0.f,0.f,0.f,0.f,0.f,0.f,0.f};
#pragma unroll
  for (int k = 0; k < ZSTR_; k += 4) {
    const int kk0 = k + hf * 2;
    const int r0  = (kk0     < LAT_) ? kk0     : 0;   // z pad col is 0
    const int r1  = (kk0 + 1 < LAT_) ? kk0 + 1 : 0;
    v2f a;  a[0]  = ap[k];                   a[1]  = ap[k + 1];
    v2f b0; b0[0] = w[(size_t)r0 * TPW_ + n_a]; b0[1] = w[(size_t)r1 * TPW_ + n_a];
    v2f b1; b1[0] = w[(size_t)r0 * TPW_ + n_b]; b1[1] = w[(size_t)r1 * TPW_ + n_b];
    acc0 = WMMA_F32(a, b0, acc0);
    acc1 = WMMA_F32(a, b1, acc1);
  }
  const float bb0 = bias[n_a];
  const float bb1 = bias[n_b];
#pragma unroll
  for (int i = 0; i < 8; ++i) {
    const int rr = mt * 16 + hf * 8 + i;
    tp[(size_t)rr * TPW_ + n_a] = tanhf(acc0[i] + bb0);
    tp[(size_t)rr * TPW_ + n_b] = tanhf(acc1[i] + bb1);
  }
}

// ---------------------------------------------------------------------------
// K7: hyperbolic part. One wave per (b, r): Mobius ops + pdist -> logits[B,4]
// ---------------------------------------------------------------------------
__global__ __launch_bounds__(256) void k_hyp(const float* __restrict__ E,
    const float* __restrict__ tp, float* __restrict__ out) {
  const int gw   = (blockIdx.x * blockDim.x + threadIdx.x) >> 5;
  const int lane = threadIdx.x & 31;
  if (gw >= B_ * NR_) return;
  const int b = gw >> 2, r = gw & 3;
  const float* e1 = E + (size_t)b * ESTR_;
  const float* e2 = e1 + EPAD_;
  const float* wu = tp + (size_t)b * TPW_ + r * MD_;
  const float* rv = wu + NR_ * MD_;
  float e2v[4], uav[4], rvv[4];
  float se2 = 0.f, sua = 0.f, sr2 = 0.f;
#pragma unroll
  for (int q = 0; q < 4; ++q) {
    const int  i = lane + q * 32;
    const bool v = (i < MD_);
    const float a1 = v ? e1[i] : 0.f;
    e2v[q] = v ? e2[i] : 0.f;
    uav[q] = (v ? wu[i] : 0.f) * a1;     // e1 * Wu
    rvv[q] = v ? rv[i] : 0.f;
    se2 += e2v[q] * e2v[q];
    sua += uav[q] * uav[q];
    sr2 += rvv[q] * rvv[q];
  }
  se2 = wave_sum(se2); sua = wave_sum(sua); sr2 = wave_sum(sr2);
  // v = expmap0(e2), u = expmap0(e1*Wu)
  const float n2  = fmaxf(sqrtf(se2), 1e-15f);
  const float tv  = tanhf(n2);
  const float scv = tv / n2;
  const float nu  = fmaxf(sqrtf(sua), 1e-15f);
  const float tu  = tanhf(nu);
  const float scu = tu / nu;
  const float v2  = scv * scv * se2;     // ||v||^2
  const float u2  = scu * scu * sua;     // ||u||^2
  float svr = 0.f;
#pragma unroll
  for (int q = 0; q < 4; ++q) svr += (scv * e2v[q]) * rvv[q];
  svr = wave_sum(svr);
  // v_m = mobius_add(v, rvh)
  const float alpha = 1.f + 2.f * svr + sr2;
  const float beta  = 1.f - v2;
  float den = 1.f + 2.f * svr + v2 * sr2;
  if (fabsf(den) < 1e-15f) den = 1e-15f;
  const float idn = 1.f / den;
  float vm[4];
  float sy2 = 0.f, sxy = 0.f;
#pragma unroll
  for (int q = 0; q < 4; ++q) {
    vm[q] = (alpha * scv * e2v[q] + beta * rvv[q]) * idn;
    sy2 += vm[q] * vm[q];
    sxy += (-(scu * uav[q])) * vm[q];
  }
  sy2 = wave_sum(sy2); sxy = wave_sum(sxy);
  // mobius_add(-u, v_m), then 2*atanh(clip(norm))
  const float a2 = 1.f + 2.f * sxy + sy2;
  const float b2 = 1.f - u2;
  float den2 = 1.f + 2.f * sxy + u2 * sy2;
  if (fabsf(den2) < 1e-15f) den2 = 1e-15f;
  float sn = 0.f;
#pragma unroll
  for (int q = 0; q < 4; ++q) {
    const float t = a2 * (-(scu * uav[q])) + b2 * vm[q];
    sn += t * t;
  }
  sn = wave_sum(sn);
  float nn = sqrtf(sn) / fabsf(den2);
  nn = fminf(nn, 1.f - 1e-5f);
  if (lane == 0) out[LOGITS_OFF + (size_t)b * NR_ + r] = 2.f * atanhf(nn);
}

// ---------------------------------------------------------------------------
// K8: rel passthrough (int bits; handles 1 or 2 32-bit words per element)
// ---------------------------------------------------------------------------
__global__ __launch_bounds__(256) void k_rel(const int* __restrict__ rel,
    float* __restrict__ out, int out_size) {
  const int b = blockIdx.x * blockDim.x + threadIdx.x;
  if (b >= B_) return;
  int* o = (int*)out;
  const int rem = out_size - REL_OFF;
  if (rem >= 2 * B_)      { o[REL_OFF + 2 * b] = rel[b]; o[REL_OFF + 2 * b + 1] = 0; }
  else if (rem >= B_)     { o[REL_OFF + b] = rel[b]; }
}

extern "C" void kernel_launch(void* const* d_in, const int* in_sizes, int n_in,
                              void* d_out, int out_size, void* d_ws, size_t ws_size,
                              hipStream_t stream) {
  const float* lm      = (const float*)d_in[0];
  const int*   ep1     = (const int*)  d_in[1];
  const int*   ep2     = (const int*)  d_in[2];
  const int*   rel     = (const int*)  d_in[3];
  const float* eps     = (const float*)d_in[4];
  const float* dropu   = (const float*)d_in[5];
  const float* w_event = (const float*)d_in[6];
  const float* b_event = (const float*)d_in[7];
  const float* w_fc1   = (const float*)d_in[8];
  const float* b_fc1   = (const float*)d_in[9];
  const float* w_mu    = (const float*)d_in[10];
  const float* b_mu    = (const float*)d_in[11];
  const float* w_sig   = (const float*)d_in[12];
  const float* b_sig   = (const float*)d_in[13];
  const float* gmu     = (const float*)d_in[14];
  const float* bmu     = (const float*)d_in[15];
  const float* gva     = (const float*)d_in[16];
  const float* bva     = (const float*)d_in[17];
  const float* w_mure  = (const float*)d_in[18];
  const float* b_mure  = (const float*)d_in[19];
  float* out = (float*)d_out;

  // workspace layout (floats)
  float* E    = (float*)d_ws;                 // B*256
  float* reps = E    + (size_t)B_ * ESTR_;    // B*300
  float* mul  = reps + (size_t)B_ * HID_;     // B*64
  float* sgl  = mul  + (size_t)B_ * ZSTR_;    // B*64
  float* z    = sgl  + (size_t)B_ * ZSTR_;    // B*64
  float* tp   = z    + (size_t)B_ * ZSTR_;    // B*800
  float* bnm  = tp   + (size_t)B_ * TPW_;     // 128
  float* bni  = bnm  + 128;                   // 128
  (void)in_sizes; (void)n_in; (void)ws_size;

  k_event  <<<128, 256, 0, stream>>>(lm, ep1, ep2, w_event, b_event, E);
  k_fc1    <<<160, 256, 0, stream>>>(E, w_fc1, b_fc1, dropu, reps);
  k_musig  <<<64,  256, 0, stream>>>(reps, w_mu, b_mu, w_sig, b_sig, mul, sgl);
  k_bnstats<<<100, 256, 0, stream>>>(mul, sgl, bnm, bni);
  k_zfin   <<<512, 256, 0, stream>>>(mul, sgl, bnm, bni, gmu, bmu, gva, bva, eps, out, z);
  k_mure   <<<400, 256, 0, stream>>>(z, w_mure, b_mure, tp);
  k_hyp    <<<1024,256, 0, stream>>>(E, tp, out);
  k_rel    <<<8,   256, 0, stream>>>(rel, out, out_size);
}